// NTM_44865228374279
// MI455X (gfx1250) — compile-verified
//
#include <hip/hip_runtime.h>
#include <math.h>

#define B_  256
#define N_  2048
#define M_  64
#define IN_ 128
#define OUT_ 64
#define H_  100
#define EPSF 1e-8f

typedef __attribute__((ext_vector_type(2))) float v2f;
typedef __attribute__((ext_vector_type(8))) float v8f;

__device__ __forceinline__ float softplusf(float x) {
    return x > 20.f ? x : log1pf(expf(x));
}
__device__ __forceinline__ float sigmoidf(float x) {
    return 1.f / (1.f + expf(-x));
}

// ---------------------------------------------------------------------------
// Kernel 1: h = x @ Wc + bc   ([256,128] x [128,100]) via fp32 WMMA 16x16x4.
// One wave (32 threads) per 16x16 output tile. K stepped by 4 (32 WMMAs).
// A frag: row = lane&15 (M), K = (lane>>4)*2 + j
// B frag: col = lane&15 (N), K = (lane>>4)*2 + j
// D frag: c[i] -> row i + (lane>=16 ? 8 : 0), col lane&15
// ---------------------------------------------------------------------------
__global__ __launch_bounds__(32) void k_h(const float* __restrict__ x,
                                          const float* __restrict__ Wc,
                                          const float* __restrict__ bc,
                                          float* __restrict__ h) {
    const int bm = blockIdx.x * 16;        // batch tile
    const int hn = blockIdx.y * 16;        // H tile (7 tiles cover 100->112)
    const int lane = threadIdx.x;
    const int fr = lane & 15;
    const int kb = (lane >> 4) * 2;
    const int col = hn + fr;
    const int ccol = col < H_ ? col : H_ - 1;          // clamp (masked below)
    const float cmask = col < H_ ? 1.f : 0.f;

    v8f c = {};
    for (int k0 = 0; k0 < IN_; k0 += 4) {
        v2f a, b;
        a.x = x[(bm + fr) * IN_ + k0 + kb];
        a.y = x[(bm + fr) * IN_ + k0 + kb + 1];
        b.x = Wc[(k0 + kb) * H_ + ccol] * cmask;
        b.y = Wc[(k0 + kb + 1) * H_ + ccol] * cmask;
        c = __builtin_amdgcn_wmma_f32_16x16x4_f32(false, a, false, b,
                                                  (short)0, c, false, false);
    }
    if (col < H_) {
        const float bcv = bc[col];
        const int rbase = (lane >= 16) ? 8 : 0;
        for (int i = 0; i < 8; ++i)
            h[(bm + i + rbase) * H_ + col] = c[i] + bcv;
    }
}

// ---------------------------------------------------------------------------
// Kernel 2: all per-batch head parameters from h[b,:].
// scal layout per b (12): [beta_r,gamma_r,g_r,s_r0..2, beta_w,gamma_w,g_w,s_w0..2]
// ---------------------------------------------------------------------------
__global__ __launch_bounds__(128) void k_params(
    const float* __restrict__ h,
    const float* rWb, const float* rbb, const float* rWgam, const float* rbgam,
    const float* rWgate, const float* rbgate, const float* rWs, const float* rbs,
    const float* rWk, const float* rbk,
    const float* wWb, const float* wbb, const float* wWgam, const float* wbgam,
    const float* wWgate, const float* wbgate, const float* wWs, const float* wbs,
    const float* wWk, const float* wbk,
    const float* wWe, const float* wbe, const float* wWa, const float* wba,
    float* __restrict__ k_r, float* __restrict__ k_w,
    float* __restrict__ e_v, float* __restrict__ a_v,
    float* __restrict__ kn_r, float* __restrict__ kn_w,
    float* __restrict__ scal) {
    __shared__ float hs[H_];
    __shared__ float kbuf[128];
    const int b = blockIdx.x, t = threadIdx.x;
    if (t < H_) hs[t] = h[b * H_ + t];
    __syncthreads();

    if (t < 64) {
        float skr = rbk[t], skw = wbk[t], se = wbe[t], sa = wba[t];
        for (int j = 0; j < H_; ++j) {
            const float hv = hs[j];
            skr += hv * rWk[j * 64 + t];
            skw += hv * wWk[j * 64 + t];
            se  += hv * wWe[j * 64 + t];
            sa  += hv * wWa[j * 64 + t];
        }
        k_r[b * 64 + t] = skr;
        k_w[b * 64 + t] = skw;
        e_v[b * 64 + t] = sigmoidf(se);
        a_v[b * 64 + t] = sa;
        kbuf[t]      = skr * skr;
        kbuf[64 + t] = skw * skw;
    } else if (t < 76) {
        const int id = t - 64;  // 0..11
        const float* W; const float* bb; int ncol, colx;
        switch (id) {
            case 0:  W = rWb;    bb = rbb;    ncol = 1; colx = 0;      break;
            case 1:  W = rWgam;  bb = rbgam;  ncol = 1; colx = 0;      break;
            case 2:  W = rWgate; bb = rbgate; ncol = 1; colx = 0;      break;
            case 3: case 4: case 5:
                     W = rWs;    bb = rbs;    ncol = 3; colx = id - 3; break;
            case 6:  W = wWb;    bb = wbb;    ncol = 1; colx = 0;      break;
            case 7:  W = wWgam;  bb = wbgam;  ncol = 1; colx = 0;      break;
            case 8:  W = wWgate; bb = wbgate; ncol = 1; colx = 0;      break;
            default: W = wWs;    bb = wbs;    ncol = 3; colx = id - 9; break;
        }
        float acc = bb[colx];
        for (int j = 0; j < H_; ++j) acc += hs[j] * W[j * ncol + colx];
        float v;
        if (id == 0 || id == 6)      v = softplusf(acc);
        else if (id == 1 || id == 7) v = 1.f + softplusf(acc);
        else if (id == 2 || id == 8) v = sigmoidf(acc);
        else                         v = acc;
        scal[b * 12 + id] = v;
    }
    __syncthreads();
    for (int s = 32; s > 0; s >>= 1) {
        if (t < s)                 kbuf[t]      += kbuf[t + s];
        if (t >= 64 && t < 64 + s) kbuf[t]      += kbuf[t + s];
        __syncthreads();
    }
    if (t == 0) kn_r[b] = fmaxf(sqrtf(kbuf[0]),  EPSF);
    if (t == 1) kn_w[b] = fmaxf(sqrtf(kbuf[64]), EPSF);
}

// ---------------------------------------------------------------------------
// Kernel 3a: cos[b,n] vs raw memory_bias (write head). 16 lanes per row,
// float4 streaming, shfl_xor width-16 reduction. Grid (B, N/16), block 256.
// ---------------------------------------------------------------------------
__global__ __launch_bounds__(256) void k_cos_plain(
    const float* __restrict__ memb, const float* __restrict__ kvec,
    const float* __restrict__ kn, float* __restrict__ cosout) {
    __shared__ float ks[64];
    const int b = blockIdx.x, t = threadIdx.x;
    if (t < 64) ks[t] = kvec[b * 64 + t];
    __syncthreads();
    const int grp = t >> 4, j = t & 15;
    const int n = blockIdx.y * 16 + grp;
    const float4 mv = ((const float4*)(memb + (((size_t)b * N_ + n) * M_)))[j];
    const float4 kv = ((const float4*)ks)[j];
    float dot = mv.x * kv.x + mv.y * kv.y + mv.z * kv.z + mv.w * kv.w;
    float ss  = mv.x * mv.x + mv.y * mv.y + mv.z * mv.z + mv.w * mv.w;
    for (int m = 8; m >= 1; m >>= 1) {
        dot += __shfl_xor(dot, m, 16);
        ss  += __shfl_xor(ss,  m, 16);
    }
    if (j == 0) {
        const float mn = fmaxf(sqrtf(ss), EPSF);
        cosout[(size_t)b * N_ + n] = dot / (kn[b] * mn);
    }
}

// ---------------------------------------------------------------------------
// Kernel 3b: cos[b,n] against the UPDATED memory, recomputed on the fly:
// mem = memb*(1 - w*e) + w*a   (read head). Never materializes mem.
// ---------------------------------------------------------------------------
__global__ __launch_bounds__(256) void k_cos_mod(
    const float* __restrict__ memb, const float* __restrict__ kvec,
    const float* __restrict__ kn, const float* __restrict__ wat,
    const float* __restrict__ e_v, const float* __restrict__ a_v,
    float* __restrict__ cosout) {
    __shared__ float ks[64], es[64], av_s[64];
    const int b = blockIdx.x, t = threadIdx.x;
    if (t < 64) {
        ks[t]   = kvec[b * 64 + t];
        es[t]   = e_v[b * 64 + t];
        av_s[t] = a_v[b * 64 + t];
    }
    __syncthreads();
    const int grp = t >> 4, j = t & 15;
    const int n = blockIdx.y * 16 + grp;
    const float w = wat[(size_t)b * N_ + n];
    const float4 mv = ((const float4*)(memb + (((size_t)b * N_ + n) * M_)))[j];
    const float4 kv = ((const float4*)ks)[j];
    const float4 ev = ((const float4*)es)[j];
    const float4 avv = ((const float4*)av_s)[j];
    float4 mm;
    mm.x = mv.x * (1.f - w * ev.x) + w * avv.x;
    mm.y = mv.y * (1.f - w * ev.y) + w * avv.y;
    mm.z = mv.z * (1.f - w * ev.z) + w * avv.z;
    mm.w = mv.w * (1.f - w * ev.w) + w * avv.w;
    float dot = mm.x * kv.x + mm.y * kv.y + mm.z * kv.z + mm.w * kv.w;
    float ss  = mm.x * mm.x + mm.y * mm.y + mm.z * mm.z + mm.w * mm.w;
    for (int m = 8; m >= 1; m >>= 1) {
        dot += __shfl_xor(dot, m, 16);
        ss  += __shfl_xor(ss,  m, 16);
    }
    if (j == 0) {
        const float mn = fmaxf(sqrtf(ss), EPSF);
        cosout[(size_t)b * N_ + n] = dot / (kn[b] * mn);
    }
}

// ---------------------------------------------------------------------------
// Kernel 4: full addressing pipeline over N for one batch row:
// softmax(bias), softmax(beta*cos), interpolate, circular 3-tap conv,
// sharpen (pow gamma), normalize. One block (256 thr) per b; wg held in LDS.
// ---------------------------------------------------------------------------
__global__ __launch_bounds__(256) void k_addr(
    const float* __restrict__ cosv, const float* __restrict__ bias,
    const float* __restrict__ scal, const int head, float* __restrict__ attn) {
    __shared__ float wg[N_];
    __shared__ float red[256];
    const int b = blockIdx.x, t = threadIdx.x;
    const float beta  = scal[b * 12 + head + 0];
    const float gamma = scal[b * 12 + head + 1];
    const float g     = scal[b * 12 + head + 2];
    const float s0    = scal[b * 12 + head + 3];
    const float s1    = scal[b * 12 + head + 4];
    const float s2    = scal[b * 12 + head + 5];
    const size_t base = (size_t)b * N_;

    float m1 = -1e30f, m2 = -1e30f;
    for (int i = t; i < N_; i += 256) {
        m1 = fmaxf(m1, bias[base + i]);
        m2 = fmaxf(m2, beta * cosv[base + i]);
    }
    red[t] = m1; __syncthreads();
    for (int s = 128; s > 0; s >>= 1) { if (t < s) red[t] = fmaxf(red[t], red[t + s]); __syncthreads(); }
    const float bmax = red[0]; __syncthreads();
    red[t] = m2; __syncthreads();
    for (int s = 128; s > 0; s >>= 1) { if (t < s) red[t] = fmaxf(red[t], red[t + s]); __syncthreads(); }
    const float cmax = red[0]; __syncthreads();

    float sb = 0.f, sc = 0.f;
    for (int i = t; i < N_; i += 256) {
        sb += expf(bias[base + i] - bmax);
        sc += expf(beta * cosv[base + i] - cmax);
    }
    red[t] = sb; __syncthreads();
    for (int s = 128; s > 0; s >>= 1) { if (t < s) red[t] += red[t + s]; __syncthreads(); }
    const float bsum = red[0]; __syncthreads();
    red[t] = sc; __syncthreads();
    for (int s = 128; s > 0; s >>= 1) { if (t < s) red[t] += red[t + s]; __syncthreads(); }
    const float csum = red[0]; __syncthreads();

    for (int i = t; i < N_; i += 256) {
        const float wc = expf(beta * cosv[base + i] - cmax) / csum;
        const float a0 = expf(bias[base + i] - bmax) / bsum;
        wg[i] = g * wc + (1.f - g) * a0;
    }
    __syncthreads();

    float pv[8];
    float psum = 0.f;
    for (int k = 0; k < 8; ++k) {
        const int i  = t + k * 256;
        const int im = (i + N_ - 1) & (N_ - 1);
        const int ip = (i + 1) & (N_ - 1);
        const float tw = s0 * wg[im] + s1 * wg[i] + s2 * wg[ip];
        const float p = powf(fmaxf(tw, 0.f), gamma);
        pv[k] = p;
        psum += p;
    }
    red[t] = psum; __syncthreads();
    for (int s = 128; s > 0; s >>= 1) { if (t < s) red[t] += red[t + s]; __syncthreads(); }
    const float inv = 1.f / (EPSF + red[0]);
    for (int k = 0; k < 8; ++k) attn[base + t + k * 256] = pv[k] * inv;
}

// ---------------------------------------------------------------------------
// Kernel 5: r[b,m] = sum_n r_attn[b,n] * mem[b,n,m], mem recomputed on the fly.
// One block per b; 4 n-groups x 64 m-lanes, LDS partial reduction.
// ---------------------------------------------------------------------------
__global__ __launch_bounds__(256) void k_read(
    const float* __restrict__ memb, const float* __restrict__ rat,
    const float* __restrict__ wat, const float* __restrict__ e_v,
    const float* __restrict__ a_v, float* __restrict__ rvec) {
    __shared__ float part[256];
    const int b = blockIdx.x, t = threadIdx.x;
    const int m = t & 63, grp = t >> 6;
    const float e = e_v[b * 64 + m];
    const float a = a_v[b * 64 + m];
    const size_t base = (size_t)b * N_;
    float acc = 0.f;
    for (int n = grp; n < N_; n += 4) {
        const float w  = wat[base + n];
        const float mb = memb[(base + n) * M_ + m];
        acc += rat[base + n] * (mb * (1.f - w * e) + w * a);
    }
    part[t] = acc;
    __syncthreads();
    if (t < 64) rvec[b * 64 + m] = part[t] + part[t + 64] + part[t + 128] + part[t + 192];
}

// ---------------------------------------------------------------------------
// Kernel 6: out = sigmoid(r @ Wo + bo)  ([256,64]x[64,64]) via fp32 WMMA.
// ---------------------------------------------------------------------------
__global__ __launch_bounds__(32) void k_out(const float* __restrict__ r,
                                            const float* __restrict__ Wo,
                                            const float* __restrict__ bo,
                                            float* __restrict__ out) {
    const int bm = blockIdx.x * 16;
    const int on = blockIdx.y * 16;
    const int lane = threadIdx.x;
    const int fr = lane & 15;
    const int kb = (lane >> 4) * 2;
    v8f c = {};
    for (int k0 = 0; k0 < M_; k0 += 4) {
        v2f a, b;
        a.x = r[(bm + fr) * M_ + k0 + kb];
        a.y = r[(bm + fr) * M_ + k0 + kb + 1];
        b.x = Wo[(k0 + kb) * OUT_ + on + fr];
        b.y = Wo[(k0 + kb + 1) * OUT_ + on + fr];
        c = __builtin_amdgcn_wmma_f32_16x16x4_f32(false, a, false, b,
                                                  (short)0, c, false, false);
    }
    const float bv = bo[on + fr];
    const int rbase = (lane >= 16) ? 8 : 0;
    for (int i = 0; i < 8; ++i)
        out[(bm + i + rbase) * OUT_ + on + fr] = sigmoidf(c[i] + bv);
}

// ---------------------------------------------------------------------------
extern "C" void kernel_launch(void* const* d_in, const int* in_sizes, int n_in,
                              void* d_out, int out_size, void* d_ws, size_t ws_size,
                              hipStream_t stream) {
    const float* x    = (const float*)d_in[0];
    const float* Wc   = (const float*)d_in[1];
    const float* bc   = (const float*)d_in[2];
    const float* Wo   = (const float*)d_in[3];
    const float* bo   = (const float*)d_in[4];
    const float* rWb  = (const float*)d_in[5];
    const float* rbb  = (const float*)d_in[6];
    const float* rWgm = (const float*)d_in[7];
    const float* rbgm = (const float*)d_in[8];
    const float* rWg  = (const float*)d_in[9];
    const float* rbg  = (const float*)d_in[10];
    const float* rWs  = (const float*)d_in[11];
    const float* rbs  = (const float*)d_in[12];
    const float* rWk  = (const float*)d_in[13];
    const float* rbk  = (const float*)d_in[14];
    const float* wWb  = (const float*)d_in[15];
    const float* wbb  = (const float*)d_in[16];
    const float* wWgm = (const float*)d_in[17];
    const float* wbgm = (const float*)d_in[18];
    const float* wWg  = (const float*)d_in[19];
    const float* wbg  = (const float*)d_in[20];
    const float* wWs  = (const float*)d_in[21];
    const float* wbs  = (const float*)d_in[22];
    const float* wWk  = (const float*)d_in[23];
    const float* wbk  = (const float*)d_in[24];
    const float* wWe  = (const float*)d_in[25];
    const float* wbe  = (const float*)d_in[26];
    const float* wWa  = (const float*)d_in[27];
    const float* wba  = (const float*)d_in[28];
    const float* rAb  = (const float*)d_in[29];
    const float* wAb  = (const float*)d_in[30];
    const float* memb = (const float*)d_in[31];
    float* out = (float*)d_out;

    // workspace layout (floats)
    float* ws    = (float*)d_ws;
    float* h_buf = ws;                     // B*H        = 25600
    float* kr    = h_buf + B_ * H_;        // B*M        = 16384
    float* kw    = kr + B_ * M_;
    float* ev    = kw + B_ * M_;
    float* av    = ev + B_ * M_;
    float* knr   = av + B_ * M_;           // B
    float* knw   = knr + B_;               // B
    float* scal  = knw + B_;               // B*12
    float* cosb  = scal + B_ * 12;         // B*N        = 524288
    float* attw  = cosb + (size_t)B_ * N_; // B*N
    float* attr  = attw + (size_t)B_ * N_; // B*N
    float* rv    = attr + (size_t)B_ * N_; // B*M

    // 1. controller GEMM (WMMA fp32)
    k_h<<<dim3(B_ / 16, (H_ + 15) / 16), 32, 0, stream>>>(x, Wc, bc, h_buf);

    // 2. all head parameters
    k_params<<<B_, 128, 0, stream>>>(h_buf,
        rWb, rbb, rWgm, rbgm, rWg, rbg, rWs, rbs, rWk, rbk,
        wWb, wbb, wWgm, wbgm, wWg, wbg, wWs, wbs, wWk, wbk,
        wWe, wbe, wWa, wba,
        kr, kw, ev, av, knr, knw, scal);

    // 3. write-head cosine vs raw memory (streams 134MB, HBM-bound)
    k_cos_plain<<<dim3(B_, N_ / 16), 256, 0, stream>>>(memb, kw, knw, cosb);

    // 4. write-head addressing -> w_attn
    k_addr<<<B_, 256, 0, stream>>>(cosb, wAb, scal, /*head=*/6, attw);

    // 5. read-head cosine vs updated memory (recomputed on the fly, L2 hits)
    k_cos_mod<<<dim3(B_, N_ / 16), 256, 0, stream>>>(memb, kr, knr, attw, ev, av, cosb);

    // 6. read-head addressing -> r_attn
    k_addr<<<B_, 256, 0, stream>>>(cosb, rAb, scal, /*head=*/0, attr);

    // 7. read vector r = r_attn . mem  (mem recomputed, L2 hits)
    k_read<<<B_, 256, 0, stream>>>(memb, attr, attw, ev, av, rv);

    // 8. output GEMM + sigmoid (WMMA fp32)
    k_out<<<dim3(B_ / 16, OUT_ / 16), 32, 0, stream>>>(rv, Wo, bo, out);
}